// LGCNConv_56788057588131
// MI455X (gfx1250) — compile-verified
//
#include <hip/hip_runtime.h>
#include <math.h>

typedef __attribute__((ext_vector_type(2))) float v2f;
typedef __attribute__((ext_vector_type(8))) float v8f;

#define NF 128      // feature dims (tail of x, full dim of h/out)
#define XROW 129    // x row length (head + tail)
#define NW 127      // valid weight rows (OUT_F-1)
#define TILE_M 32   // nodes per block
#define LDSP 132    // padded LDS row stride (floats): conflict-free, 16B-aligned rows
#define NTHREADS 256

// Dynamic-LDS partition offsets (floats)
#define OFF_WS 0
#define OFF_AS (NF * LDSP)                   // 16896
#define OFF_VS (OFF_AS + TILE_M * LDSP)      // + 4224
#define OFF_BS (OFF_VS + TILE_M * LDSP)      // + 4224
#define OFF_HS (OFF_BS + NF)
#define OFF_SC (OFF_HS + TILE_M)
#define OFF_FA (OFF_SC + TILE_M)
#define OFF_HD (OFF_FA + TILE_M)
#define LDS_FLOATS (OFF_HD + TILE_M)         // 25600 floats = 102400 bytes

// Pointer types matching the async-LDS builtin signature:
// (int4 in AS1, int4 in AS3, imm offset, imm cpol)
typedef int v4i_vs __attribute__((vector_size(16)));
typedef __attribute__((address_space(1))) v4i_vs* gptr128;
typedef __attribute__((address_space(3))) v4i_vs* lptr128;

__device__ __forceinline__ void wait_async_zero() {
#if __has_builtin(__builtin_amdgcn_s_wait_asynccnt)
  __builtin_amdgcn_s_wait_asynccnt(0);
#else
  asm volatile("s_wait_asynccnt 0x0" ::: "memory");
#endif
}

// ---------------- Kernel 1: log_map_zero -> GEMM (f32 WMMA) -> exp_map_zero
__global__ __launch_bounds__(NTHREADS) void k1_transform(
    const float* __restrict__ x, const float* __restrict__ w,
    const float* __restrict__ bias, float* __restrict__ h, int n_nodes)
{
  extern __shared__ float smem[];
  float* __restrict__ Ws = smem + OFF_WS;   // [NF][LDSP] padded weight (row 127 = 0)
  float* __restrict__ As = smem + OFF_AS;   // [TILE_M][LDSP] scaled tangent tails
  float* __restrict__ Vs = smem + OFF_VS;   // [TILE_M][LDSP] v = [0, mx+bias]
  float* __restrict__ bs = smem + OFF_BS;   // [NF] padded bias
  float* __restrict__ Hs = smem + OFF_HS;
  float* __restrict__ sc = smem + OFF_SC;
  float* __restrict__ fa = smem + OFF_FA;
  float* __restrict__ hd = smem + OFF_HD;

  const int t = threadIdx.x;
  const int node0 = blockIdx.x * TILE_M;

  // ---- Stage padded weight into LDS (rows 0..126), coalesced 16B per lane.
  // Async path: ASYNCcnt-tracked DMA into LDS, overlapped with x staging +
  // log-map scale below; waited on just before the WMMA stage consumes Ws.
#if __has_builtin(__builtin_amdgcn_global_load_async_to_lds_b128)
  for (int f = t; f < (NW * NF) / 4; f += NTHREADS) {
    int j = f >> 5, k4 = (f & 31) << 2;
    __builtin_amdgcn_global_load_async_to_lds_b128(
        (gptr128)(w + f * 4), (lptr128)&Ws[j * LDSP + k4], 0, 0);
  }
#else
  const float4* __restrict__ w4 = (const float4*)w;
  for (int f = t; f < (NW * NF) / 4; f += NTHREADS) {
    int j = f >> 5, k4 = (f & 31) << 2;
    *(float4*)&Ws[j * LDSP + k4] = w4[f];
  }
#endif
  if (t < NF) {
    Ws[NW * LDSP + t] = 0.f;                 // zero pad row 127 (disjoint from async writes)
    bs[t] = (t < NW) ? bias[t] : 0.f;
  }

  // ---- Stage x tile: head -> Hs, tail -> As
  for (int idx = t; idx < TILE_M * XROW; idx += NTHREADS) {
    int m = idx / XROW, c = idx % XROW;
    int node = node0 + m;
    float v = (node < n_nodes) ? x[node * XROW + c] : 0.f;
    if (c == 0) Hs[m] = v;
    else        As[m * LDSP + (c - 1)] = v;
  }
  __syncthreads();

  // ---- Per-node log_map_zero scale = arcosh(head) / ||tail||
  if (t < TILE_M) {
    float sum = 0.f;
    for (int k = 0; k < NF; ++k) { float a = As[t * LDSP + k]; sum += a * a; }
    float tn = sqrtf(fmaxf(sum, 1e-6f));
    float z  = fmaxf(Hs[t], 1.0f + 1e-7f);
    float dist = logf(z + sqrtf(z * z - 1.0f));
    sc[t] = (node0 + t < n_nodes) ? (dist / tn) : 0.f;
    Vs[t * LDSP + 0] = 0.f;
  }
  __syncthreads();

  for (int idx = t; idx < TILE_M * NF; idx += NTHREADS) {
    int m = idx / NF, k = idx % NF;
    As[m * LDSP + k] *= sc[m];
  }
  wait_async_zero();                         // Ws fully landed in LDS before GEMM
  __syncthreads();

  // ---- WMMA GEMM: mx[m][j] = sum_k As[m][k] * Ws[j][k]
  // 16 wave-tiles = 2 row blocks x 8 column tiles, uniform over 8 waves (EXEC all 1s)
  const int wv   = t >> 5;
  const int lane = t & 31;
  const int half = lane >> 4;
  const int lm   = lane & 15;
  for (int wt = wv; wt < 16; wt += 8) {
    const int rb   = (wt >> 3) << 4;        // row block base: 0 or 16
    const int colj = ((wt & 7) << 4) + lm;  // weight row index j (tile N)
    const float bv = bs[colj];              // bias folded into accumulator init
    v8f acc = { bv, bv, bv, bv, bv, bv, bv, bv };
    #pragma unroll
    for (int kk = 0; kk < NF; kk += 4) {
      const int kb = kk + half * 2;
      v2f a, b;
      a.x = As[(rb + lm) * LDSP + kb];
      a.y = As[(rb + lm) * LDSP + kb + 1];
      b.x = Ws[colj * LDSP + kb];
      b.y = Ws[colj * LDSP + kb + 1];
      acc = __builtin_amdgcn_wmma_f32_16x16x4_f32(false, a, false, b,
                                                  (short)0, acc, false, false);
    }
    #pragma unroll
    for (int r = 0; r < 8; ++r)             // colj==127 lands in pad column (value 0)
      Vs[(rb + r + half * 8) * LDSP + 1 + colj] = acc[r];
  }
  __syncthreads();

  // ---- exp_map_zero on v
  if (t < TILE_M) {
    float sum = 0.f;
    for (int c = 1; c < NF; ++c) { float v = Vs[t * LDSP + c]; sum += v * v; }
    float nn   = sqrtf(fmaxf(sum, 1e-6f));
    float ncut = fminf(nn, 50.f);
    float fac  = sinhf(ncut) / nn;
    fa[t] = fac;
    hd[t] = sqrtf(1.0f + fac * fac * sum);
  }
  __syncthreads();

  for (int idx = t; idx < TILE_M * NF; idx += NTHREADS) {
    int m = idx / NF, c = idx % NF;
    int node = node0 + m;
    if (node < n_nodes)
      h[node * NF + c] = (c == 0) ? hd[m] : fa[m] * Vs[m * LDSP + c];
  }
}

// ---------------- Kernel 2: segment-sum centroid + hyp_act (fused)
__device__ __forceinline__ int lower_bound_i(const int* __restrict__ a, int n, int key) {
  int lo = 0, hi = n;
  while (lo < hi) { int mid = (lo + hi) >> 1; if (a[mid] < key) lo = mid + 1; else hi = mid; }
  return lo;
}

__global__ __launch_bounds__(128) void k2_agg_act(
    const float* __restrict__ h, const int* __restrict__ row,
    const int* __restrict__ colv, const float* __restrict__ val,
    int n_edges, float* __restrict__ out)
{
  __shared__ float red[128];
  __shared__ int   se[2];
  __shared__ float bc0;
  const int node = blockIdx.x;
  const int t = threadIdx.x;   // feature dim 0..127

  if (t < 2) se[t] = lower_bound_i(row, n_edges, node + t);
  __syncthreads();
  const int start = se[0], end = se[1];

  // weighted gather-sum (h is L2-resident: 25.6 MB << 192 MB)
  float s = 0.f;
  for (int e = start; e < end; ++e) {
    int c = colv[e];
    s = fmaf(val[e], h[c * NF + t], s);
  }

  // Lorentz inner = sum_{d>=1} s^2 - s0^2
  red[t] = (t == 0) ? -s * s : s * s;
  __syncthreads();
  for (int off = 64; off > 0; off >>= 1) { if (t < off) red[t] += red[t + off]; __syncthreads(); }
  float inner = red[0]; __syncthreads();
  float coef = rsqrtf(fmaxf(fabsf(inner), 1e-6f));
  float cv = coef * s;

  if (t == 0) bc0 = cv;
  __syncthreads();
  float c0 = bc0;

  // log_map_zero tail norm
  red[t] = (t == 0) ? 0.f : cv * cv;
  __syncthreads();
  for (int off = 64; off > 0; off >>= 1) { if (t < off) red[t] += red[t + off]; __syncthreads(); }
  float tsum = red[0]; __syncthreads();
  float tn = sqrtf(fmaxf(tsum, 1e-6f));
  float z  = fmaxf(c0, 1.0f + 1e-7f);
  float dist = logf(z + sqrtf(z * z - 1.0f));
  float xt = (t == 0) ? 0.f : fmaxf((dist / tn) * cv, 0.f);   // relu(log_map)

  // exp_map_zero
  red[t] = xt * xt;
  __syncthreads();
  for (int off = 64; off > 0; off >>= 1) { if (t < off) red[t] += red[t + off]; __syncthreads(); }
  float xsum = red[0]; __syncthreads();
  float nn   = sqrtf(fmaxf(xsum, 1e-6f));
  float ncut = fminf(nn, 50.f);
  float fac  = sinhf(ncut) / nn;
  float head = sqrtf(1.0f + fac * fac * xsum);
  out[node * NF + t] = (t == 0) ? head : fac * xt;
}

extern "C" void kernel_launch(void* const* d_in, const int* in_sizes, int n_in,
                              void* d_out, int out_size, void* d_ws, size_t ws_size,
                              hipStream_t stream) {
  const float* x    = (const float*)d_in[0];
  const float* w    = (const float*)d_in[1];
  const float* bias = (const float*)d_in[2];
  const float* val  = (const float*)d_in[3];
  const int*   row  = (const int*)d_in[4];
  const int*   col  = (const int*)d_in[5];
  float* out = (float*)d_out;
  float* h   = (float*)d_ws;   // 50000*128*4 = 25.6 MB intermediate

  const int n_nodes = in_sizes[0] / XROW;
  const int n_edges = in_sizes[4];

  const size_t lds_bytes = LDS_FLOATS * sizeof(float);   // 102400 B (<= 320 KB WGP LDS)
  k1_transform<<<dim3((n_nodes + TILE_M - 1) / TILE_M), dim3(NTHREADS), lds_bytes, stream>>>(
      x, w, bias, h, n_nodes);
  k2_agg_act<<<dim3(n_nodes), dim3(128), 0, stream>>>(
      h, row, col, val, n_edges, out);
}